// BinomialLoss_73469710566012
// MI455X (gfx1250) — compile-verified
//
#include <hip/hip_runtime.h>
#include <stdint.h>

// Problem constants (fixed by the reference): 8192x8192 sim matrix, 128 classes.
#define N_DIM 8192
#define BLOCK 256
#define VEC   4
#define SWEEP (N_DIM / (BLOCK * VEC))   // 8 sweeps of 256 lanes * float4

#define ALPHA  40.0f
#define BETA   2.0f
#define MARGIN 0.5f

typedef float vf4 __attribute__((ext_vector_type(4)));
typedef int   vi4 __attribute__((ext_vector_type(4)));
// Type used by the async-to-LDS builtin signature (b128 = 4 x i32).
typedef int   v4i __attribute__((vector_size(16)));

#define AS_G __attribute__((address_space(1)))
#define AS_L __attribute__((address_space(3)))

// CDNA5 async global->LDS copy path (ASYNCcnt-tracked); probe-confirmed to
// lower to global_load_async_to_lds_b128 on this toolchain.
#if defined(__has_builtin)
#if __has_builtin(__builtin_amdgcn_global_load_async_to_lds_b128) && \
    __has_builtin(__builtin_amdgcn_s_wait_asynccnt)
#define USE_ASYNC_LDS 1
#endif
#endif
#ifndef USE_ASYNC_LDS
#define USE_ASYNC_LDS 0
#endif

// Generic->LDS pointer: low 32 bits of a generic LDS address are the
// wave-relative LDS byte offset.
__device__ __forceinline__ AS_L v4i* to_lds_v4(const void* p) {
  return (AS_L v4i*)(uint32_t)(uintptr_t)p;
}
__device__ __forceinline__ AS_G v4i* to_global_v4(const void* p) {
  return (AS_G v4i*)(uintptr_t)p;
}

__global__ __launch_bounds__(BLOCK) void BinomialLoss_73469710566012_kernel(
    const float* __restrict__ sim, const int* __restrict__ tgt,
    float* __restrict__ out_loss, float* __restrict__ out_grad) {
  // 32 KB row cache + 8 B counters: 4+ workgroups fit per 320 KB WGP LDS pool.
  __shared__ __align__(16) float s_row[N_DIM];
  __shared__ int s_cnt[2];  // [0] = pos count, [1] = neg count

  const int row = blockIdx.x;
  const int tid = threadIdx.x;
  const float* grow = sim + (size_t)row * N_DIM;

  if (tid < 2) s_cnt[tid] = 0;

  // ---- Stage 1: async-copy this row into LDS (single HBM read of sim) ----
#if USE_ASYNC_LDS
#pragma unroll
  for (int k = 0; k < SWEEP; ++k) {
    const int j = (tid + k * BLOCK) * VEC;  // 512B contiguous per wave
    __builtin_amdgcn_global_load_async_to_lds_b128(
        to_global_v4(grow + j), to_lds_v4(&s_row[j]), 0, 0);
  }
  __builtin_amdgcn_s_wait_asynccnt(0);
#else
#pragma unroll
  for (int k = 0; k < SWEEP; ++k) {
    const int j = (tid + k * BLOCK) * VEC;
    vf4 v = __builtin_nontemporal_load((const vf4*)(grow + j));
    *(vf4*)&s_row[j] = v;
  }
#endif
  __syncthreads();

  // ---- Stage 2: per-row pos/neg counts; pack predicates into bitmasks ----
  const int ti = tgt[row];  // uniform scalar load, L2-hot
  uint32_t same_mask = 0u;  // bit (k*4+c): targets match
  uint32_t pos_mask = 0u;   // bit (k*4+c): match && sim < 1
#pragma unroll
  for (int k = 0; k < SWEEP; ++k) {
    const int j = (tid + k * BLOCK) * VEC;
    vf4 x = *(const vf4*)&s_row[j];
    vi4 t = *(const vi4*)(tgt + j);  // 32KB targets stay resident in L2
#pragma unroll
    for (int c = 0; c < VEC; ++c) {
      const uint32_t bit = 1u << (k * VEC + c);
      const bool same = (t[c] == ti);
      same_mask |= same ? bit : 0u;
      pos_mask |= (same && (x[c] < 1.0f)) ? bit : 0u;
    }
  }
  atomicAdd(&s_cnt[0], (int)__popc(pos_mask));       // ds_add_u32
  atomicAdd(&s_cnt[1], (int)(32u - __popc(same_mask)));
  __syncthreads();

  const int posCnt = s_cnt[0];
  const int negCnt = s_cnt[1];
  const float valid = (negCnt > 0) ? 1.0f : 0.0f;  // rows w/o negatives -> 0
  const float ipos = valid / (float)(posCnt > 0 ? posCnt : 1);
  const float ineg = valid / (float)(negCnt > 0 ? negCnt : 1);

  // ---- Stage 3: branchless loss/grad from LDS; NT streaming stores ----
  // Unified form: with c = same ? -BETA : ALPHA and a = c*(x - m):
  //   loss = softplus(a) * mask,   grad = c * sigmoid(a) * inv_count * mask
  // One v_exp shared: e = exp(-|a|) -> softplus(a) = max(a,0) + log(1+e),
  //                                    sigmoid(a) = (a>=0 ? 1 : e) / (1+e).
  float* lrow = out_loss + (size_t)row * N_DIM;
  float* grow_o = out_grad + (size_t)row * N_DIM;
#pragma unroll
  for (int k = 0; k < SWEEP; ++k) {
    const int j = (tid + k * BLOCK) * VEC;
    vf4 x = *(const vf4*)&s_row[j];
    vf4 l4, g4;
#pragma unroll
    for (int c = 0; c < VEC; ++c) {
      const int idx = k * VEC + c;
      const bool same = (same_mask >> idx) & 1u;
      const bool pos = (pos_mask >> idx) & 1u;

      const float cf = same ? -BETA : ALPHA;
      const float a = cf * (x[c] - MARGIN);

      const float e = __expf(-fabsf(a));                 // v_exp_f32 (abs mod)
      const float onepe = 1.0f + e;
      const float r = __builtin_amdgcn_rcpf(onepe);      // v_rcp_f32
      const float sp = fmaxf(a, 0.0f) + __logf(onepe);   // v_log_f32
      const float sig = (a >= 0.0f) ? r : e * r;

      const float lm = same ? (pos ? valid : 0.0f) : valid;
      const float gm = same ? (pos ? ipos : 0.0f) : ineg;

      l4[c] = sp * lm;
      g4[c] = cf * sig * gm;
    }
    // 512 MB of pure streaming output: NT hint keeps L2 for sim/targets.
    __builtin_nontemporal_store(l4, (vf4*)(lrow + j));
    __builtin_nontemporal_store(g4, (vf4*)(grow_o + j));
  }
}

extern "C" void kernel_launch(void* const* d_in, const int* in_sizes, int n_in,
                              void* d_out, int out_size, void* d_ws, size_t ws_size,
                              hipStream_t stream) {
  (void)in_sizes; (void)n_in; (void)out_size; (void)d_ws; (void)ws_size;
  const float* sim = (const float*)d_in[0];
  const int* tgt = (const int*)d_in[1];
  float* out_loss = (float*)d_out;
  float* out_grad = out_loss + (size_t)N_DIM * N_DIM;
  BinomialLoss_73469710566012_kernel<<<dim3(N_DIM), dim3(BLOCK), 0, stream>>>(
      sim, tgt, out_loss, out_grad);
}